// FixedNet2_56040733278665
// MI455X (gfx1250) — compile-verified
//
#include <hip/hip_runtime.h>
#include <hip/hip_bf16.h>
#include <math.h>

// ---------------------------------------------------------------------------
// FixedNet2 GNN on MI455X (gfx1250, wave32)
//   L0: a0 = x@w_neigh0 ; base0 = b0 + x@w_self0          (WMMA f32 16x16x4)
//   edge: agg[dst] += a[src]*ew                            (f32 atomics, x3)
//   L1/L2: h = relu(agg+base); a = h*wn; base = b + h*ws   (scalar streams)
//   graph segment-sum -> 64x1 ; tiny MLP head + log_softmax
// Bandwidth-bound (~630 MB/call); fp32 throughout for exactness.
// Edge arrays (src+dst+ew = 192 MB) match the 192 MB L2: passes 2-3 can hit
// L2, so default (RT) temporal hints are used for them on purpose.
// ---------------------------------------------------------------------------

typedef __attribute__((ext_vector_type(2))) float v2f;
typedef __attribute__((ext_vector_type(8))) float v8f;

// ---- init: zero agg[N] and hg[G] -----------------------------------------
__global__ __launch_bounds__(256) void init_ws(float* __restrict__ agg,
                                               float* __restrict__ hg,
                                               int N, int G) {
  int i = blockIdx.x * 256 + threadIdx.x;
  if (i < N) agg[i] = 0.0f;
  if (i < G) hg[i]  = 0.0f;
}

// ---- layer-0 projection via V_WMMA_F32_16X16X4_F32 ------------------------
// Per wave: one 16-node tile.  A = x[tile] (16x16, consumed as 4 K=4 chunks),
// B = [w_neigh0 | w_self0 | 0...] (16x16).  D col 0 -> a_out, col 1 -> base.
// B is built branch-free: unconditional cached weight loads + v_cndmask
// selects, so EXEC stays all-1s without save/restore churn around the WMMAs.
__global__ __launch_bounds__(256) void proj_wmma(
    const float* __restrict__ x,
    const float* __restrict__ wn,   // w_neigh0 [16]
    const float* __restrict__ wsf,  // w_self0  [16]
    const float* __restrict__ b0,   // bias     [1]
    float* __restrict__ a_out,
    float* __restrict__ base_out,
    int nTiles, int N) {
  const int lane = threadIdx.x & 31;
  const int wave = threadIdx.x >> 5;
  const int tile = blockIdx.x * 8 + wave;   // 8 waves / 256-thread block

  if (tile < nTiles) {                       // wave-uniform: EXEC all-1s inside
    const int col = lane & 15;
    const int hi  = lane >> 4;               // 0: lanes 0-15, 1: lanes 16-31
    const bool isC0 = (col == 0);
    const bool isC1 = (col == 1);
    const float* xrow = x + (size_t)(tile * 16 + col) * 16;

    // Preload weight pairs for all 4 K-chunks (unconditional, cache-hot),
    // then select per-lane column without branching.
    v2f Bv[4];
#pragma unroll
    for (int c4 = 0; c4 < 4; ++c4) {
      const int k0 = c4 * 4 + 2 * hi;
      const float wn0v = wn[k0],  wn1v = wn[k0 + 1];
      const float ws0v = wsf[k0], ws1v = wsf[k0 + 1];
      Bv[c4].x = isC0 ? wn0v : (isC1 ? ws0v : 0.0f);
      Bv[c4].y = isC0 ? wn1v : (isC1 ? ws1v : 0.0f);
    }

    v8f c = {0.f, 0.f, 0.f, 0.f, 0.f, 0.f, 0.f, 0.f};
#pragma unroll
    for (int c4 = 0; c4 < 4; ++c4) {
      const int k0 = c4 * 4 + 2 * hi;        // ISA A layout: v0=K(k0), v1=K(k0+1)
      v2f A;
      A.x = xrow[k0];
      A.y = xrow[k0 + 1];
      c = __builtin_amdgcn_wmma_f32_16x16x4_f32(false, A, false, Bv[c4],
                                                (short)0, c, false, false);
    }

    const float bias = b0[0];
    // D layout: reg r holds M = r + 8*hi for column n = lane&15
    if (isC0) {
#pragma unroll
      for (int r = 0; r < 8; ++r) a_out[tile * 16 + hi * 8 + r] = c[r];
    } else if (isC1) {
#pragma unroll
      for (int r = 0; r < 8; ++r) base_out[tile * 16 + hi * 8 + r] = c[r] + bias;
    }
  }

  // scalar tail for N % 16 (zero for N=500000, kept for generality)
  int rem = N - nTiles * 16;
  if (blockIdx.x == 0 && (int)threadIdx.x < rem) {
    int n = nTiles * 16 + threadIdx.x;
    const float* xr = x + (size_t)n * 16;
    float s0 = 0.f, s1 = 0.f;
    for (int k = 0; k < 16; ++k) { s0 += xr[k] * wn[k]; s1 += xr[k] * wsf[k]; }
    a_out[n] = s0;
    base_out[n] = s1 + b0[0];
  }
}

// ---- edge aggregation: agg[dst] += a[src] * ew ----------------------------
// 4 edges/thread: 3x 128-bit streaming loads; gathers/atomics hit L2 (2MB set).
__global__ __launch_bounds__(256) void edge_agg(
    const int* __restrict__ src, const int* __restrict__ dst,
    const float* __restrict__ ew, const float* __restrict__ a,
    float* __restrict__ agg, int E) {
  int gid = blockIdx.x * 256 + threadIdx.x;
  int quads = E >> 2;
  if (gid < quads) {
    int4   s = ((const int4*)src)[gid];
    int4   d = ((const int4*)dst)[gid];
    float4 w = ((const float4*)ew)[gid];
    atomicAdd(&agg[d.x], a[s.x] * w.x);
    atomicAdd(&agg[d.y], a[s.y] * w.y);
    atomicAdd(&agg[d.z], a[s.z] * w.z);
    atomicAdd(&agg[d.w], a[s.w] * w.w);
  }
  int t = (quads << 2) + gid;                // tail (E%4 edges)
  if (t < E) atomicAdd(&agg[dst[t]], a[src[t]] * ew[t]);
}

// ---- node update between layers (scalar 1->1 weights) ---------------------
// h = relu(agg+base); a = h*wn; base = b + h*ws; agg = 0 (ready for next pass)
__global__ __launch_bounds__(256) void node_update(
    float* __restrict__ agg, float* __restrict__ base, float* __restrict__ a,
    const float* __restrict__ wn, const float* __restrict__ b,
    const float* __restrict__ wsf, int N) {
  int i = blockIdx.x * 256 + threadIdx.x;
  if (i < N) {
    float h = agg[i] + base[i];
    h = h > 0.f ? h : 0.f;
    a[i]    = h * wn[0];
    base[i] = b[0] + h * wsf[0];
    agg[i]  = 0.f;
  }
}

// ---- final relu + per-graph segment sum (LDS histogram, then atomics) -----
__global__ __launch_bounds__(256) void node_final(
    const float* __restrict__ agg, const float* __restrict__ base,
    const int* __restrict__ gid, float* __restrict__ hg, int N, int G) {
  __shared__ float sh[256];
  sh[threadIdx.x] = 0.0f;
  __syncthreads();
  int i = blockIdx.x * 256 + threadIdx.x;
  bool lds_path = (G <= 256);
  if (i < N) {
    float h = agg[i] + base[i];
    h = h > 0.f ? h : 0.f;
    int g = gid[i];
    if (lds_path) atomicAdd(&sh[g], h);      // ds_add_f32; graph_id sorted ->
    else          atomicAdd(&hg[g], h);      // ~1-2 live bins per block
  }
  __syncthreads();
  if (lds_path && (int)threadIdx.x < G) {
    float v = sh[threadIdx.x];
    if (v != 0.0f) atomicAdd(&hg[threadIdx.x], v);
  }
}

// ---- head: fc1 -> sigmoid(z*1000) -> out -> relu -> log_softmax -----------
__global__ __launch_bounds__(64) void head_kernel(
    const float* __restrict__ hg, const float* __restrict__ fc1w,
    const float* __restrict__ fc1b, const float* __restrict__ outw,
    const float* __restrict__ outb, float* __restrict__ out, int G) {
  int g = blockIdx.x * 64 + threadIdx.x;
  if (g >= G) return;
  float v = hg[g];
  float t[8];
#pragma unroll
  for (int j = 0; j < 8; ++j) {
    float z = (v * fc1w[j] + fc1b[j]) * 1000.0f;
    t[j] = 1.0f / (1.0f + expf(-z));
  }
  float o[4], m = -INFINITY;
#pragma unroll
  for (int k = 0; k < 4; ++k) {
    float s = outb[k];
#pragma unroll
    for (int j = 0; j < 8; ++j) s += t[j] * outw[k * 8 + j];
    o[k] = s > 0.f ? s : 0.f;
    m = fmaxf(m, o[k]);
  }
  float se = 0.f;
#pragma unroll
  for (int k = 0; k < 4; ++k) se += expf(o[k] - m);
  float lse = m + logf(se);
#pragma unroll
  for (int k = 0; k < 4; ++k) out[g * 4 + k] = o[k] - lse;
}

// ---------------------------------------------------------------------------
extern "C" void kernel_launch(void* const* d_in, const int* in_sizes, int n_in,
                              void* d_out, int out_size, void* d_ws,
                              size_t ws_size, hipStream_t stream) {
  const float* x    = (const float*)d_in[0];
  const int*   src  = (const int*)d_in[1];
  const int*   dst  = (const int*)d_in[2];
  const int*   gid  = (const int*)d_in[3];
  const float* ew   = (const float*)d_in[4];
  // d_in[5] = num_graphs (device scalar) — G derived host-side from out_size
  const float* wn0  = (const float*)d_in[6];
  const float* b0   = (const float*)d_in[7];
  const float* ws0  = (const float*)d_in[8];
  const float* wn1  = (const float*)d_in[9];
  const float* b1   = (const float*)d_in[10];
  const float* ws1  = (const float*)d_in[11];
  const float* wn2  = (const float*)d_in[12];
  const float* b2   = (const float*)d_in[13];
  const float* ws2  = (const float*)d_in[14];
  const float* fc1w = (const float*)d_in[15];
  const float* fc1b = (const float*)d_in[16];
  const float* outw = (const float*)d_in[17];
  const float* outb = (const float*)d_in[18];
  float* out = (float*)d_out;

  const int N = in_sizes[0] / 16;
  const int E = in_sizes[1];
  const int G = out_size / 4;

  float* a    = (float*)d_ws;     // [N]
  float* base = a + N;            // [N]
  float* agg  = base + N;         // [N]
  float* hg   = agg + N;          // [G]

  const int nb    = (N + 255) / 256;
  const int tiles = N / 16;
  const int pb    = tiles > 0 ? (tiles + 7) / 8 : 1;
  const int quads = E / 4;
  const int eb    = quads > 0 ? (quads + 255) / 256 : 1;

  init_ws<<<nb, 256, 0, stream>>>(agg, hg, N, G);
  proj_wmma<<<pb, 256, 0, stream>>>(x, wn0, ws0, b0, a, base, tiles, N);

  edge_agg<<<eb, 256, 0, stream>>>(src, dst, ew, a, agg, E);        // layer 0
  node_update<<<nb, 256, 0, stream>>>(agg, base, a, wn1, b1, ws1, N);
  edge_agg<<<eb, 256, 0, stream>>>(src, dst, ew, a, agg, E);        // layer 1
  node_update<<<nb, 256, 0, stream>>>(agg, base, a, wn2, b2, ws2, N);
  edge_agg<<<eb, 256, 0, stream>>>(src, dst, ew, a, agg, E);        // layer 2
  node_final<<<nb, 256, 0, stream>>>(agg, base, gid, hg, N, G);

  head_kernel<<<(G + 63) / 64, 64, 0, stream>>>(hg, fc1w, fc1b, outw, outb,
                                                out, G);
}